// GFNO_56427280335235
// MI455X (gfx1250) — compile-verified
//
#include <hip/hip_runtime.h>
#include <math.h>

#define BB 8
#define NNODE 1024
#define TIN 10
#define TOUT 20
#define WID 32
#define KS 16
#define MTIME 4
#define KNNB 8
#define NITER 30
#define NTP (NNODE*TOUT)   // 20480

typedef __attribute__((ext_vector_type(2))) float v2f;
typedef __attribute__((ext_vector_type(8))) float v8f;

__device__ __forceinline__ v8f wmma4(v2f a, v2f b, v8f c) {
  // V_WMMA_F32_16X16X4_F32 : D = A(16x4) * B(4x16) + C(16x16), all f32
  return __builtin_amdgcn_wmma_f32_16x16x4_f32(false, a, false, b, (short)0, c, false, false);
}

__device__ __forceinline__ float gelu_f(float x) {
  return 0.5f * x * (1.0f + erff(x * 0.7071067811865475f));
}

// ---------------- stage 0: lifting MLP (13 -> 32) ----------------
__global__ void k_lift(const float* __restrict__ x, const float* __restrict__ pos,
                       const float* __restrict__ Wp, const float* __restrict__ bp,
                       float* __restrict__ h) {
  int idx = blockIdx.x * blockDim.x + threadIdx.x;
  if (idx >= BB * NNODE * TOUT) return;
  int t = idx % TOUT;
  int n = (idx / TOUT) % NNODE;
  int b = idx / (TOUT * NNODE);
  float in[13];
  for (int i = 0; i < TIN; i++) in[i] = x[((size_t)b * NNODE + n) * TIN + i];
  in[10] = pos[((size_t)b * NNODE + n) * 2 + 0];
  in[11] = pos[((size_t)b * NNODE + n) * 2 + 1];
  in[12] = (float)(TIN + t) / (float)(TOUT + TIN - 1);
  for (int c = 0; c < WID; c++) {
    float a = bp[c];
    for (int i = 0; i < 13; i++) a += in[i] * Wp[i * WID + c];
    h[(((size_t)b * WID + c) * NNODE + n) * TOUT + t] = a;
  }
}

// ---------------- per-node squared norms of features at t=0 ----------------
__global__ void k_sq(const float* __restrict__ h, float* __restrict__ sq) {
  int idx = blockIdx.x * blockDim.x + threadIdx.x;
  if (idx >= BB * NNODE) return;
  int b = idx / NNODE, n = idx % NNODE;
  float s = 0.f;
  for (int c = 0; c < WID; c++) {
    float v = h[(((size_t)b * WID + c) * NNODE + n) * TOUT];
    s += v * v;
  }
  sq[idx] = s;
}

// ---------------- Gram via WMMA -> pairwise distance matrix ----------------
__global__ void k_gram_dm(const float* __restrict__ h, const float* __restrict__ sq,
                          float* __restrict__ Dm) {
  int lane = threadIdx.x & 31;
  int wv = blockIdx.x * (blockDim.x >> 5) + (threadIdx.x >> 5);
  const int tiles = (NNODE / 16) * (NNODE / 16);
  if (wv >= BB * tiles) return;
  int b = wv / tiles, rem = wv % tiles;
  int it = rem / (NNODE / 16), jt = rem % (NNODE / 16);
  int hf = lane >> 4, lr = lane & 15;
  int i0 = it * 16, j0 = jt * 16;
  const float* Xp = h + (size_t)b * WID * NNODE * TOUT;  // Xp[n][c] = h[b][c][n][0]
  v8f acc = {0.f, 0.f, 0.f, 0.f, 0.f, 0.f, 0.f, 0.f};
  for (int kk = 0; kk < WID; kk += 4) {
    int c0 = kk + 2 * hf;
    v2f a, bm;
    a.x = Xp[(size_t)(c0) * NNODE * TOUT + (size_t)(i0 + lr) * TOUT];
    a.y = Xp[(size_t)(c0 + 1) * NNODE * TOUT + (size_t)(i0 + lr) * TOUT];
    bm.x = Xp[(size_t)(c0) * NNODE * TOUT + (size_t)(j0 + lr) * TOUT];
    bm.y = Xp[(size_t)(c0 + 1) * NNODE * TOUT + (size_t)(j0 + lr) * TOUT];
    acc = wmma4(a, bm, acc);
  }
  for (int r = 0; r < 8; r++) {
    int i = i0 + 8 * hf + r, j = j0 + lr;
    float g = acc[r];
    float d2 = sq[b * NNODE + i] + sq[b * NNODE + j] - 2.0f * g;
    d2 = fmaxf(d2, 0.0f);
    Dm[((size_t)b * NNODE + i) * NNODE + j] = (d2 > 0.0f) ? sqrtf(d2) : 0.0f;
  }
}

// ---------------- sigma = mean(Dm) per batch ----------------
__global__ void k_sigma(const float* __restrict__ Dm, float* __restrict__ sigma) {
  __shared__ float red[256];
  int b = blockIdx.x;
  float s = 0.f;
  for (size_t e = threadIdx.x; e < (size_t)NNODE * NNODE; e += 256)
    s += Dm[(size_t)b * NNODE * NNODE + e];
  red[threadIdx.x] = s;
  __syncthreads();
  for (int st = 128; st > 0; st >>= 1) {
    if ((int)threadIdx.x < st) red[threadIdx.x] += red[threadIdx.x + st];
    __syncthreads();
  }
  if (threadIdx.x == 0) sigma[b] = red[0] / (float)((size_t)NNODE * NNODE);
}

// ---------------- kNN (8 smallest incl. self) ----------------
__global__ void k_knn(const float* __restrict__ Dm, const float* __restrict__ sigma,
                      float* __restrict__ kval, int* __restrict__ kidx) {
  int idx = blockIdx.x * blockDim.x + threadIdx.x;
  if (idx >= BB * NNODE) return;
  int b = idx / NNODE, n = idx % NNODE;
  float d8[KNNB];
  int i8[KNNB];
  for (int i = 0; i < KNNB; i++) { d8[i] = 3.4e38f; i8[i] = 0; }
  const float* row = Dm + ((size_t)b * NNODE + n) * NNODE;
  for (int j = 0; j < NNODE; j++) {
    float d = row[j];
    if (d < d8[KNNB - 1]) {
      int p = KNNB - 1;
      while (p > 0 && d8[p - 1] > d) { d8[p] = d8[p - 1]; i8[p] = i8[p - 1]; p--; }
      d8[p] = d; i8[p] = j;
    }
  }
  float sg = sigma[b];
  float inv = 1.0f / (sg * sg);
  for (int i = 0; i < KNNB; i++) {
    kval[(size_t)idx * KNNB + i] = expf(-d8[i] * inv);
    kidx[(size_t)idx * KNNB + i] = i8[i];
  }
}

__global__ void k_zero(float* __restrict__ p, size_t n) {
  for (size_t e = (size_t)blockIdx.x * blockDim.x + threadIdx.x; e < n;
       e += (size_t)gridDim.x * blockDim.x)
    p[e] = 0.0f;
}

// A = 0.5*(A_knn + A_knn^T)  (two commutative float atomics per edge -> deterministic)
__global__ void k_scatter(const float* __restrict__ kval, const int* __restrict__ kidx,
                          float* __restrict__ A) {
  int idx = blockIdx.x * blockDim.x + threadIdx.x;
  if (idx >= BB * NNODE * KNNB) return;
  int b = idx / (NNODE * KNNB);
  int n = (idx / KNNB) % NNODE;
  int j = kidx[idx];
  float v = 0.5f * kval[idx];
  atomicAdd(&A[((size_t)b * NNODE + n) * NNODE + j], v);
  atomicAdd(&A[((size_t)b * NNODE + j) * NNODE + n], v);
}

__global__ void k_deg(const float* __restrict__ A, float* __restrict__ deg) {
  int idx = blockIdx.x * blockDim.x + threadIdx.x;
  if (idx >= BB * NNODE) return;
  const float* row = A + (size_t)idx * NNODE;
  float s = 0.f;
  for (int j = 0; j < NNODE; j++) s += row[j];
  deg[idx] = s;
}

// M = 2I - Dinv (deg*I - A) Dinv   (overwrites the Dm buffer)
__global__ void k_buildM(const float* __restrict__ A, const float* __restrict__ deg,
                         float* __restrict__ M) {
  size_t e = (size_t)blockIdx.x * blockDim.x + threadIdx.x;
  if (e >= (size_t)BB * NNODE * NNODE) return;
  size_t b = e / ((size_t)NNODE * NNODE);
  size_t rem = e % ((size_t)NNODE * NNODE);
  int i = (int)(rem / NNODE), j = (int)(rem % NNODE);
  float di = rsqrtf(deg[b * NNODE + i] + 1e-6f);
  float dj = rsqrtf(deg[b * NNODE + j] + 1e-6f);
  float l = di * (((i == j) ? deg[b * NNODE + i] : 0.0f) - A[e]) * dj;
  M[e] = ((i == j) ? 2.0f : 0.0f) - l;
}

__global__ void k_initX(float* __restrict__ X) {
  int idx = blockIdx.x * blockDim.x + threadIdx.x;
  if (idx >= BB * NNODE * KS) return;
  unsigned u = (unsigned)idx * 2654435761u + 12345u;
  u ^= u >> 13; u *= 2246822519u; u ^= u >> 16;
  X[idx] = (float)(u & 0xFFFFu) / 65536.0f - 0.5f;
}

// ---------------- Y = M * X  via WMMA ([1024x1024]x[1024x16]) ----------------
__global__ void k_mxv(const float* __restrict__ M, const float* __restrict__ X,
                      float* __restrict__ Y) {
  int lane = threadIdx.x & 31;
  int wv = blockIdx.x * (blockDim.x >> 5) + (threadIdx.x >> 5);
  if (wv >= BB * (NNODE / 16)) return;
  int b = wv / (NNODE / 16), it = wv % (NNODE / 16);
  int hf = lane >> 4, lr = lane & 15;
  const float* Mb = M + (size_t)b * NNODE * NNODE;
  const float* Xb = X + (size_t)b * NNODE * KS;
  v8f acc = {0.f, 0.f, 0.f, 0.f, 0.f, 0.f, 0.f, 0.f};
  for (int kk = 0; kk < NNODE; kk += 4) {
    int k0 = kk + 2 * hf;
    v2f a, bm;
    a.x = Mb[(size_t)(it * 16 + lr) * NNODE + k0];
    a.y = Mb[(size_t)(it * 16 + lr) * NNODE + k0 + 1];
    bm.x = Xb[(size_t)k0 * KS + lr];
    bm.y = Xb[(size_t)(k0 + 1) * KS + lr];
    acc = wmma4(a, bm, acc);
  }
  for (int r = 0; r < 8; r++)
    Y[(size_t)b * NNODE * KS + (size_t)(it * 16 + 8 * hf + r) * KS + lr] = acc[r];
}

// ---------------- S = A^T B  (A,B : [N,16]) , one block per batch ----------------
__global__ void k_gram16(const float* __restrict__ Am, const float* __restrict__ Bm,
                         float* __restrict__ S) {
  __shared__ float ta[64][16];
  __shared__ float tb[64][16];
  int b = blockIdx.x;
  int tid = threadIdx.x;
  int k1 = tid >> 4, k2 = tid & 15;
  float acc = 0.f;
  for (int n0 = 0; n0 < NNODE; n0 += 64) {
    for (int e = tid; e < 64 * 16; e += 256) {
      ta[e >> 4][e & 15] = Am[(size_t)b * NNODE * KS + (size_t)(n0 + (e >> 4)) * KS + (e & 15)];
      tb[e >> 4][e & 15] = Bm[(size_t)b * NNODE * KS + (size_t)(n0 + (e >> 4)) * KS + (e & 15)];
    }
    __syncthreads();
    for (int n = 0; n < 64; n++) acc += ta[n][k1] * tb[n][k2];
    __syncthreads();
  }
  S[b * 256 + tid] = acc;
}

// ---------------- Cholesky of 16x16 S, output R^{-1} (upper) ----------------
__global__ void k_cholinv(const float* __restrict__ S, float* __restrict__ Rinv) {
  if (threadIdx.x != 0) return;
  int b = blockIdx.x;
  float Aa[16][16], R[16][16], Q[16][16];
  for (int i = 0; i < 16; i++)
    for (int j = 0; j < 16; j++) {
      Aa[i][j] = S[b * 256 + i * 16 + j];
      R[i][j] = 0.f; Q[i][j] = 0.f;
    }
  for (int i = 0; i < 16; i++) Aa[i][i] += 1e-6f;
  for (int j = 0; j < 16; j++) {
    float s = Aa[j][j];
    for (int k = 0; k < j; k++) s -= R[k][j] * R[k][j];
    float d = sqrtf(fmaxf(s, 1e-12f));
    R[j][j] = d;
    for (int i = j + 1; i < 16; i++) {
      float t = Aa[j][i];
      for (int k = 0; k < j; k++) t -= R[k][j] * R[k][i];
      R[j][i] = t / d;
    }
  }
  for (int j = 0; j < 16; j++) {
    Q[j][j] = 1.0f / R[j][j];
    for (int i = j - 1; i >= 0; i--) {
      float t = 0.f;
      for (int k = i + 1; k <= j; k++) t += R[i][k] * Q[k][j];
      Q[i][j] = -t / R[i][i];
    }
  }
  for (int i = 0; i < 16; i++)
    for (int j = 0; j < 16; j++) Rinv[b * 256 + i * 16 + j] = Q[i][j];
}

// ---------------- Out[n][k] = sum_j In[n][j] * Mt[j][k] ----------------
__global__ void k_apply16(const float* __restrict__ In, const float* __restrict__ Mt,
                          float* __restrict__ Out) {
  int idx = blockIdx.x * blockDim.x + threadIdx.x;
  if (idx >= BB * NNODE * KS) return;
  int b = idx / (NNODE * KS);
  int rem = idx % (NNODE * KS);
  int n = rem / KS, k = rem % KS;
  float acc = 0.f;
  for (int j = 0; j < KS; j++)
    acc += In[(size_t)b * NNODE * KS + (size_t)n * KS + j] * Mt[b * 256 + j * 16 + k];
  Out[idx] = acc;
}

// ---------------- 16x16 Jacobi eigensolver, columns sorted by eig desc(M) ----------------
__global__ void k_jacobi16(const float* __restrict__ S, float* __restrict__ V) {
  if (threadIdx.x != 0) return;
  int b = blockIdx.x;
  float Aa[16][16], Vv[16][16];
  for (int i = 0; i < 16; i++)
    for (int j = 0; j < 16; j++) {
      Aa[i][j] = 0.5f * (S[b * 256 + i * 16 + j] + S[b * 256 + j * 16 + i]);
      Vv[i][j] = (i == j) ? 1.0f : 0.0f;
    }
  for (int sw = 0; sw < 16; sw++) {
    for (int p = 0; p < 15; p++)
      for (int q = p + 1; q < 16; q++) {
        float apq = Aa[p][q];
        if (fabsf(apq) < 1e-10f) continue;
        float app = Aa[p][p], aqq = Aa[q][q];
        float th = 0.5f * (aqq - app) / apq;
        float tt = ((th >= 0.f) ? 1.0f : -1.0f) / (fabsf(th) + sqrtf(th * th + 1.0f));
        float c = rsqrtf(tt * tt + 1.0f), s = tt * c;
        for (int i = 0; i < 16; i++) {
          float aip = Aa[i][p], aiq = Aa[i][q];
          Aa[i][p] = c * aip - s * aiq;
          Aa[i][q] = s * aip + c * aiq;
        }
        for (int j = 0; j < 16; j++) {
          float apj = Aa[p][j], aqj = Aa[q][j];
          Aa[p][j] = c * apj - s * aqj;
          Aa[q][j] = s * apj + c * aqj;
        }
        for (int i = 0; i < 16; i++) {
          float vip = Vv[i][p], viq = Vv[i][q];
          Vv[i][p] = c * vip - s * viq;
          Vv[i][q] = s * vip + c * viq;
        }
      }
  }
  float ev[16];
  for (int i = 0; i < 16; i++) ev[i] = Aa[i][i];
  // descending in mu(M) == ascending in lambda(L) = 2 - mu
  for (int k = 0; k < 16; k++) {
    int best = k;
    for (int j = k + 1; j < 16; j++)
      if (ev[j] > ev[best]) best = j;
    if (best != k) {
      float t = ev[k]; ev[k] = ev[best]; ev[best] = t;
      for (int i = 0; i < 16; i++) {
        float tv = Vv[i][k]; Vv[i][k] = Vv[i][best]; Vv[i][best] = tv;
      }
    }
  }
  for (int i = 0; i < 16; i++)
    for (int j = 0; j < 16; j++) V[b * 256 + i * 16 + j] = Vv[i][j];
}

// ---------------- xs[k][t] = sum_n basis[n][k] * h[c][n][t]  via WMMA ----------------
__global__ void k_proj(const float* __restrict__ h, const float* __restrict__ basis,
                       float* __restrict__ xs) {
  int lane = threadIdx.x & 31;
  int wv = blockIdx.x * (blockDim.x >> 5) + (threadIdx.x >> 5);
  if (wv >= BB * WID * 2) return;
  int b = wv / (WID * 2), rem = wv % (WID * 2);
  int c = rem / 2, tt = rem % 2;
  int hf = lane >> 4, lr = lane & 15, t0 = tt * 16;
  const float* hc = h + ((size_t)b * WID + c) * NNODE * TOUT;
  const float* bs = basis + (size_t)b * NNODE * KS;
  int t = t0 + lr;
  v8f acc = {0.f, 0.f, 0.f, 0.f, 0.f, 0.f, 0.f, 0.f};
  for (int kk = 0; kk < NNODE; kk += 4) {
    int n0 = kk + 2 * hf;
    v2f a, bm;
    a.x = bs[(size_t)n0 * KS + lr];
    a.y = bs[(size_t)(n0 + 1) * KS + lr];
    bm.x = (t < TOUT) ? hc[(size_t)n0 * TOUT + t] : 0.0f;
    bm.y = (t < TOUT) ? hc[(size_t)(n0 + 1) * TOUT + t] : 0.0f;
    acc = wmma4(a, bm, acc);
  }
  for (int r = 0; r < 8; r++) {
    int k = 8 * hf + r;
    if (t < TOUT) xs[(((size_t)b * WID + c) * KS + k) * TOUT + t] = acc[r];
  }
}

// ---------------- truncated rDFT over T=20 (modes 0..3) ----------------
__global__ void k_rfft(const float* __restrict__ xs, float* __restrict__ xfr,
                       float* __restrict__ xfi) {
  int idx = blockIdx.x * blockDim.x + threadIdx.x;
  if (idx >= BB * WID * KS * MTIME) return;
  int m = idx & 3;
  int bck = idx >> 2;
  const float* p = xs + (size_t)bck * TOUT;
  float re = 0.f, im = 0.f;
  for (int t = 0; t < TOUT; t++) {
    float ang = -6.283185307179586f * (float)(m * t) / (float)TOUT;
    re += p[t] * cosf(ang);
    im += p[t] * sinf(ang);
  }
  xfr[idx] = re;
  xfi[idx] = im;
}

// ---------------- complex channel mix per (k,m) ----------------
__global__ void k_cmix(const float* __restrict__ xfr, const float* __restrict__ xfi,
                       const float* __restrict__ fwr, const float* __restrict__ fwi,
                       float* __restrict__ ofr, float* __restrict__ ofi, int layer) {
  int idx = blockIdx.x * blockDim.x + threadIdx.x;
  if (idx >= BB * WID * KS * MTIME) return;
  int m = idx & 3;
  int k = (idx >> 2) & 15;
  int o = (idx >> 6) & 31;
  int b = idx >> 11;
  float ar = 0.f, ai = 0.f;
  for (int i = 0; i < WID; i++) {
    int xin = (((b * WID + i) * KS + k) << 2) + m;
    float xr = xfr[xin], xi = xfi[xin];
    size_t w = ((((size_t)layer * WID + i) * WID + o) * KS + k) * MTIME + m;
    float wr = fwr[w], wi = fwi[w];
    ar += xr * wr - xi * wi;
    ai += xr * wi + xi * wr;
  }
  ofr[idx] = ar;
  ofi[idx] = ai;
}

// ---------------- inverse rDFT back to T=20 ----------------
__global__ void k_irfft(const float* __restrict__ ofr, const float* __restrict__ ofi,
                        float* __restrict__ os) {
  int idx = blockIdx.x * blockDim.x + threadIdx.x;
  if (idx >= BB * WID * KS * TOUT) return;
  int t = idx % TOUT;
  int bck = idx / TOUT;
  float s = ofr[(size_t)bck * MTIME];
  for (int m = 1; m < MTIME; m++) {
    float th = 6.283185307179586f * (float)(m * t) / (float)TOUT;
    s += 2.0f * (ofr[(size_t)bck * MTIME + m] * cosf(th) - ofi[(size_t)bck * MTIME + m] * sinf(th));
  }
  os[idx] = s * (1.0f / (float)TOUT);
}

// ---------------- 1x1 conv h2 = cw*h + cb via WMMA ----------------
__global__ void k_point(const float* __restrict__ h, const float* __restrict__ cw,
                        const float* __restrict__ cb, float* __restrict__ hn, int layer) {
  int lane = threadIdx.x & 31;
  int wv = blockIdx.x * (blockDim.x >> 5) + (threadIdx.x >> 5);
  const int ptile = NTP / 16;  // 1280
  if (wv >= BB * 2 * ptile) return;
  int b = wv / (2 * ptile), rem = wv % (2 * ptile);
  int mt = rem / ptile, pt = rem % ptile;
  int hf = lane >> 4, lr = lane & 15;
  int o0 = mt * 16, p0 = pt * 16;
  const float* cwl = cw + (size_t)layer * WID * WID;
  v8f acc = {0.f, 0.f, 0.f, 0.f, 0.f, 0.f, 0.f, 0.f};
  for (int kk = 0; kk < WID; kk += 4) {
    int i0 = kk + 2 * hf;
    v2f a, bm;
    a.x = cwl[(o0 + lr) * WID + i0];
    a.y = cwl[(o0 + lr) * WID + i0 + 1];
    bm.x = h[((size_t)b * WID + i0) * NTP + p0 + lr];
    bm.y = h[((size_t)b * WID + i0 + 1) * NTP + p0 + lr];
    acc = wmma4(a, bm, acc);
  }
  for (int r = 0; r < 8; r++) {
    int o = o0 + 8 * hf + r;
    hn[((size_t)b * WID + o) * NTP + p0 + lr] = acc[r] + cb[layer * WID + o];
  }
}

// ---------------- hn += basis * os ; gelu for layers 0..2 ----------------
__global__ void k_combine(float* __restrict__ hn, const float* __restrict__ basis,
                          const float* __restrict__ os, int layer) {
  int idx = blockIdx.x * blockDim.x + threadIdx.x;
  if (idx >= BB * WID * NNODE * TOUT) return;
  int t = idx % TOUT;
  int n = (idx / TOUT) % NNODE;
  int c = (idx / (TOUT * NNODE)) % WID;
  int b = idx / (TOUT * NNODE * WID);
  float v = hn[idx];
  for (int k = 0; k < KS; k++)
    v += basis[((size_t)b * NNODE + n) * KS + k] *
         os[(((size_t)b * WID + c) * KS + k) * TOUT + t];
  if (layer < 3) v = gelu_f(v);
  hn[idx] = v;
}

// ---------------- head: (h@q1w+q1b) -> gelu -> @q2w+q2b, fused WMMA ----------------
__global__ void k_head(const float* __restrict__ h, const float* __restrict__ q1w,
                       const float* __restrict__ q1b, const float* __restrict__ q2w,
                       const float* __restrict__ q2b, float* __restrict__ out) {
  int lane = threadIdx.x & 31;
  int wv = blockIdx.x * (blockDim.x >> 5) + (threadIdx.x >> 5);
  const int RT = BB * NNODE * TOUT;  // 163840
  if (wv >= RT / 16) return;
  int hf = lane >> 4, lr = lane & 15;
  int rbase = wv * 16;
  int r = rbase + lr;
  int b = r / (NNODE * TOUT);
  int rem = r % (NNODE * TOUT);
  int n = rem / TOUT, t = rem % TOUT;
  v2f areg[8];
  for (int s = 0; s < 8; s++) {
    int c0 = 4 * s + 2 * hf;
    areg[s].x = h[(((size_t)b * WID + c0) * NNODE + n) * TOUT + t];
    areg[s].y = h[(((size_t)b * WID + c0 + 1) * NNODE + n) * TOUT + t];
  }
  float partial[8];
  for (int i = 0; i < 8; i++) partial[i] = 0.f;
  for (int nt = 0; nt < 8; nt++) {
    v8f acc = {0.f, 0.f, 0.f, 0.f, 0.f, 0.f, 0.f, 0.f};
    for (int s = 0; s < 8; s++) {
      int c0 = 4 * s + 2 * hf;
      v2f bm;
      bm.x = q1w[c0 * 128 + nt * 16 + lr];
      bm.y = q1w[(c0 + 1) * 128 + nt * 16 + lr];
      acc = wmma4(areg[s], bm, acc);
    }
    for (int rr = 0; rr < 8; rr++) {
      int col = nt * 16 + lr;
      float g = gelu_f(acc[rr] + q1b[col]);
      partial[rr] += g * q2w[col];
    }
  }
  for (int rr = 0; rr < 8; rr++) {
    float v = partial[rr];
    for (int m = 8; m >= 1; m >>= 1) v += __shfl_xor(v, m, 32);
    if (lr == 0) out[rbase + 8 * hf + rr] = v + q2b[0];
  }
}

extern "C" void kernel_launch(void* const* d_in, const int* in_sizes, int n_in,
                              void* d_out, int out_size, void* d_ws, size_t ws_size,
                              hipStream_t stream) {
  (void)in_sizes; (void)n_in; (void)out_size; (void)ws_size;
  const float* x = (const float*)d_in[0];
  const float* pos = (const float*)d_in[1];
  const float* Wp = (const float*)d_in[2];
  const float* bp = (const float*)d_in[3];
  const float* fwr = (const float*)d_in[4];
  const float* fwi = (const float*)d_in[5];
  const float* cw = (const float*)d_in[6];
  const float* cb = (const float*)d_in[7];
  const float* q1w = (const float*)d_in[8];
  const float* q1b = (const float*)d_in[9];
  const float* q2w = (const float*)d_in[10];
  const float* q2b = (const float*)d_in[11];
  float* out = (float*)d_out;

  float* ws = (float*)d_ws;
  size_t off = 0;
  float* hA = ws + off;   off += (size_t)BB * WID * NNODE * TOUT;
  float* hB = ws + off;   off += (size_t)BB * WID * NNODE * TOUT;
  float* big1 = ws + off; off += (size_t)BB * NNODE * NNODE;  // Dm, then M
  float* big2 = ws + off; off += (size_t)BB * NNODE * NNODE;  // A
  float* sq = ws + off;   off += (size_t)BB * NNODE;
  float* sigma = ws + off; off += BB;
  float* kval = ws + off; off += (size_t)BB * NNODE * KNNB;
  int* kidx = (int*)(ws + off); off += (size_t)BB * NNODE * KNNB;
  float* deg = ws + off;  off += (size_t)BB * NNODE;
  float* Xb = ws + off;   off += (size_t)BB * NNODE * KS;
  float* Yb = ws + off;   off += (size_t)BB * NNODE * KS;
  float* Sb = ws + off;   off += (size_t)BB * 256;
  float* Rb = ws + off;   off += (size_t)BB * 256;
  float* Vb = ws + off;   off += (size_t)BB * 256;
  float* xs = ws + off;   off += (size_t)BB * WID * KS * TOUT;
  float* xfr = ws + off;  off += (size_t)BB * WID * KS * MTIME;
  float* xfi = ws + off;  off += (size_t)BB * WID * KS * MTIME;
  float* ofr = ws + off;  off += (size_t)BB * WID * KS * MTIME;
  float* ofi = ws + off;  off += (size_t)BB * WID * KS * MTIME;
  float* osb = ws + off;  off += (size_t)BB * WID * KS * TOUT;

  dim3 blk(256);
  k_lift<<<(BB * NNODE * TOUT + 255) / 256, blk, 0, stream>>>(x, pos, Wp, bp, hA);

  float* cur = hA;
  float* nxt = hB;
  for (int layer = 0; layer < 4; layer++) {
    k_sq<<<(BB * NNODE + 255) / 256, blk, 0, stream>>>(cur, sq);
    k_gram_dm<<<(BB * 64 * 64) / 8, blk, 0, stream>>>(cur, sq, big1);
    k_sigma<<<BB, blk, 0, stream>>>(big1, sigma);
    k_knn<<<(BB * NNODE + 255) / 256, blk, 0, stream>>>(big1, sigma, kval, kidx);
    k_zero<<<8192, blk, 0, stream>>>(big2, (size_t)BB * NNODE * NNODE);
    k_scatter<<<(BB * NNODE * KNNB + 255) / 256, blk, 0, stream>>>(kval, kidx, big2);
    k_deg<<<(BB * NNODE + 255) / 256, blk, 0, stream>>>(big2, deg);
    k_buildM<<<(int)(((size_t)BB * NNODE * NNODE + 255) / 256), blk, 0, stream>>>(big2, deg, big1);
    k_initX<<<(BB * NNODE * KS + 255) / 256, blk, 0, stream>>>(Xb);
    for (int it = 0; it < NITER; it++) {
      k_mxv<<<(BB * (NNODE / 16)) / 8, blk, 0, stream>>>(big1, Xb, Yb);
      k_gram16<<<BB, blk, 0, stream>>>(Yb, Yb, Sb);
      k_cholinv<<<BB, 64, 0, stream>>>(Sb, Rb);
      k_apply16<<<(BB * NNODE * KS + 255) / 256, blk, 0, stream>>>(Yb, Rb, Xb);
    }
    // Rayleigh-Ritz: T = X^T (M X), rotate X by eigvecs of T
    k_mxv<<<(BB * (NNODE / 16)) / 8, blk, 0, stream>>>(big1, Xb, Yb);
    k_gram16<<<BB, blk, 0, stream>>>(Xb, Yb, Sb);
    k_jacobi16<<<BB, 64, 0, stream>>>(Sb, Vb);
    k_apply16<<<(BB * NNODE * KS + 255) / 256, blk, 0, stream>>>(Xb, Vb, Yb);  // basis -> Yb

    k_proj<<<(BB * WID * 2) / 8, blk, 0, stream>>>(cur, Yb, xs);
    k_rfft<<<(BB * WID * KS * MTIME + 255) / 256, blk, 0, stream>>>(xs, xfr, xfi);
    k_cmix<<<(BB * WID * KS * MTIME + 255) / 256, blk, 0, stream>>>(xfr, xfi, fwr, fwi, ofr, ofi, layer);
    k_irfft<<<(BB * WID * KS * TOUT + 255) / 256, blk, 0, stream>>>(ofr, ofi, osb);
    k_point<<<(BB * 2 * (NTP / 16)) / 8, blk, 0, stream>>>(cur, cw, cb, nxt, layer);
    k_combine<<<(BB * WID * NNODE * TOUT + 255) / 256, blk, 0, stream>>>(nxt, Yb, osb, layer);
    float* tmp = cur; cur = nxt; nxt = tmp;
  }
  k_head<<<((BB * NNODE * TOUT) / 16) / 8, blk, 0, stream>>>(cur, q1w, q1b, q2w, q2b, out);
}